// DirectionNet_84138409328764
// MI455X (gfx1250) — compile-verified
//
#include <hip/hip_runtime.h>
#include <hip/hip_bf16.h>

typedef __attribute__((ext_vector_type(2))) float v2f;
typedef __attribute__((ext_vector_type(8))) float v8f;

#define NB    8       // batch
#define HW    256     // downsampled H and W
#define HIN   2048    // input H and W
#define TAPS  81      // 9x9 conv taps
#define KPAD  84      // taps padded to multiple of 4 (WMMA K-step = 4)
#define ND    5       // reduced direction channels

// workspace layout (in floats)
#define WS_XD  0              // xd: 8*256*256      = 524288 floats
#define WS_WR  524288         // Wr: 5*84           = 420 floats
#define WS_RED 524800         // reduced: 8*5*256^2 = 2621440 floats

// ---------------------------------------------------------------------------
// Kernel 1: 8x8 area pool (== F.interpolate scale 1/8 'area').
// Pure streaming read of 134 MB -> the roofline-dominant stage. float4 loads,
// one thread per output pixel (16 x b128 loads, fully coalesced per wave).
// ---------------------------------------------------------------------------
__global__ void k_avgpool(const float* __restrict__ x, float* __restrict__ xd) {
    int idx = blockIdx.x * blockDim.x + threadIdx.x;   // 524288 threads
    int b  = idx >> 16;
    int yo = (idx >> 8) & 255;
    int xo = idx & 255;
    const float* p = x + (((size_t)b * HIN) + (size_t)yo * 8) * HIN + xo * 8;
    float s = 0.f;
#pragma unroll
    for (int r = 0; r < 8; ++r) {
        const float4* q = (const float4*)(p + (size_t)r * HIN);
        float4 a = q[0], c = q[1];
        s += (a.x + a.y) + (a.z + a.w) + (c.x + c.y) + (c.z + c.w);
    }
    xd[idx] = s * 0.015625f;   // 1/64
}

// ---------------------------------------------------------------------------
// Kernel 2: fold the constant 15->5 reduction matrix into the conv weights:
// Wr[d][t] = sum_{j in bucket(d)} weight[j][t].  Bucket membership is the
// compile-time constant structure of build_reduce_matrix() (bitmask per d).
// Taps 81..83 are zero (K padding).
// ---------------------------------------------------------------------------
__global__ void k_weights(const float* __restrict__ w, float* __restrict__ Wr) {
    const unsigned mask[ND] = {0x0001u, 0x0038u, 0x03C0u, 0x1C00u, 0x6007u};
    for (int i = threadIdx.x; i < ND * KPAD; i += blockDim.x) {
        int d = i / KPAD, t = i - d * KPAD;
        float s = 0.f;
        if (t < TAPS) {
            unsigned mk = mask[d];
            for (int j = 0; j < 15; ++j)
                if ((mk >> j) & 1u) s += w[j * TAPS + t];
        }
        Wr[i] = s;
    }
}

// ---------------------------------------------------------------------------
// Kernel 3: implicit-GEMM 9x9 conv via V_WMMA_F32_16X16X4_F32.
//   per wave: D(16px x 16ch) += A(16px x 4taps) * B(4taps x 16ch), 21 k-steps.
// Block = 256 threads = 8 waves = 128 output pixels of one row y, one batch.
// xd halo patch (10x136, row 9 = zero pad so the k-loop is branch-free and
// EXEC stays all-ones across every WMMA) and zero-padded 16x84 B tile in LDS.
// A-frag layout: lane L -> M=L%16, VGPR0/1 -> K = (L<16?0:2)+{0,1}.
// B-frag layout mirrors A with N=L%16. D: VGPR r -> M = r + (L<16?0:8).
// ---------------------------------------------------------------------------
__global__ void k_conv_wmma(const float* __restrict__ xd,
                            const float* __restrict__ Wr,
                            float* __restrict__ red) {
    __shared__ float sX[10 * 136];      // 9 live rows + 1 zero pad row
    __shared__ float sB[16 * KPAD];

    const int b     = blockIdx.z;
    const int y     = blockIdx.y;
    const int xbase = blockIdx.x * 128;
    const int tid   = threadIdx.x;

    for (int i = tid; i < 10 * 136; i += 256) {
        int a = i / 136, j = i - a * 136;
        int gy = y + a - 4, gx = xbase + j - 4;
        float v = 0.f;
        if (a < 9 && (unsigned)gy < HW && (unsigned)gx < HW)
            v = xd[((size_t)b * HW + gy) * HW + gx];
        sX[i] = v;
    }
    for (int i = tid; i < 16 * KPAD; i += 256) {
        int n = i / KPAD;
        sB[i] = (n < ND) ? Wr[i - n * KPAD + n * KPAD] : 0.f;  // == Wr[i] for n<5
    }
    __syncthreads();

    const int wave = tid >> 5;
    const int lane = tid & 31;
    const int m    = lane & 15;
    const int hi   = lane >> 4;         // 0 or 1
    const int k0   = hi * 2;            // K base for this half-wave
    const int xl   = wave * 16 + m;     // pixel within the 128-wide chunk
    const float* bp = sB + m * KPAD;    // N = lane & 15 (zeros beyond ch 4/tap 80)

    v8f c = {0.f, 0.f, 0.f, 0.f, 0.f, 0.f, 0.f, 0.f};
#pragma unroll
    for (int kk = 0; kk < 21; ++kk) {
        int t0 = kk * 4 + k0;
        int t1 = t0 + 1;
        v2f av, bv;
        av.x = sX[(t0 / 9) * 136 + xl + (t0 % 9)];   // tap>=81 reads zero row 9
        av.y = sX[(t1 / 9) * 136 + xl + (t1 % 9)];
        bv.x = bp[t0];
        bv.y = bp[t1];
        // (neg_a, A, neg_b, B, c_mod, C, reuse_a, reuse_b)
        c = __builtin_amdgcn_wmma_f32_16x16x4_f32(false, av, false, bv,
                                                  (short)0, c, false, false);
    }

    if (m < ND) {                        // only channels 0..4 are real
        int px = xbase + wave * 16 + hi * 8;
        float* o = red + (((size_t)b * ND + m) * HW + y) * HW + px;
        *(float4*)o       = make_float4(c[0], c[1], c[2], c[3]);
        *(float4*)(o + 4) = make_float4(c[4], c[5], c[6], c[7]);
    }
}

// ---------------------------------------------------------------------------
// Kernel 4: 3x3 stride-1 max-pool (clamped window == -inf padding), banker's
// round (rintf, RNE default == jnp.round), multiply by broadcast xd.
// reduced (10.5 MB) is L2-resident on MI455X (192 MB L2).
// ---------------------------------------------------------------------------
__global__ void k_maxpool_mul(const float* __restrict__ red,
                              const float* __restrict__ xd,
                              float* __restrict__ out) {
    int idx = blockIdx.x * blockDim.x + threadIdx.x;
    if (idx >= NB * ND * HW * HW) return;
    int x  = idx & 255;
    int y  = (idx >> 8) & 255;
    int bd = idx >> 16;                 // b*5 + d
    int b  = bd / ND;
    const float* r = red + (size_t)bd * HW * HW;
    int y0 = y > 0 ? y - 1 : 0,  y1 = y < HW - 1 ? y + 1 : HW - 1;
    int x0 = x > 0 ? x - 1 : 0,  x1 = x < HW - 1 ? x + 1 : HW - 1;
    float mx = -3.402823466e38f;
    for (int yy = y0; yy <= y1; ++yy)
        for (int xx = x0; xx <= x1; ++xx)
            mx = fmaxf(mx, r[yy * HW + xx]);
    out[idx] = rintf(mx) * xd[(size_t)b * (HW * HW) + y * HW + x];
}

// ---------------------------------------------------------------------------
extern "C" void kernel_launch(void* const* d_in, const int* in_sizes, int n_in,
                              void* d_out, int out_size, void* d_ws, size_t ws_size,
                              hipStream_t stream) {
    const float* x = (const float*)d_in[0];   // [8,1,2048,2048]
    const float* w = (const float*)d_in[1];   // [15,1,9,9]
    float* ws  = (float*)d_ws;
    float* xd  = ws + WS_XD;
    float* Wr  = ws + WS_WR;
    float* red = ws + WS_RED;
    float* out = (float*)d_out;               // [8,5,256,256]

    k_avgpool    <<<2048, 256, 0, stream>>>(x, xd);
    k_weights    <<<1, 128, 0, stream>>>(w, Wr);
    k_conv_wmma  <<<dim3(2, HW, NB), 256, 0, stream>>>(xd, Wr, red);
    k_maxpool_mul<<<(NB * ND * HW * HW + 255) / 256, 256, 0, stream>>>(red, xd, out);
}